// TFAdaptiveEmbedding_35613868819141
// MI455X (gfx1250) — compile-verified
//
#include <hip/hip_runtime.h>

typedef float v2f __attribute__((ext_vector_type(2)));
typedef float v8f __attribute__((ext_vector_type(8)));

#define D_PROJ 1024
#define GLOBAL_AS __attribute__((address_space(1)))
typedef const GLOBAL_AS float* gcfloat_p;
typedef GLOBAL_AS float*       gfloat_p;
typedef const GLOBAL_AS int*   gcint_p;

struct GemmParams {
  const float* emb0; const float* emb1; const float* emb2; const float* emb3;
  const float* proj0; const float* proj1; const float* proj2; const float* proj3;
  const int* inp;
  float* out;
  int* counts;
  int* lists;
  int nTok;
};

__global__ void init_counts_kernel(int* counts) {
  if (threadIdx.x < 4) counts[threadIdx.x] = 0;
}

__global__ void classify_kernel(const int* __restrict__ inp, int nTok,
                                int* __restrict__ counts, int* __restrict__ lists) {
  int t = blockIdx.x * blockDim.x + threadIdx.x;
  if (t >= nTok) return;
  int v = ((gcint_p)inp)[t];
  int bk = (v < 20000) ? 0 : (v < 40000) ? 1 : (v < 200000) ? 2 : 3;
  int slot = atomicAdd(&counts[bk], 1);
  ((GLOBAL_AS int*)lists)[bk * nTok + slot] = t;
}

__launch_bounds__(256)
__global__ void gemm_kernel(GemmParams p) {
  __shared__ int stok[16];   // token flat positions of this m-tile (-1 = invalid)
  __shared__ int sloc[16];   // local row index within the bucket's table
  extern __shared__ float ldsA[];   // 16 x (K+2) f32, padded rows (bank-conflict free)

  const int bk    = blockIdx.y;
  const int cnt   = ((gcint_p)p.counts)[bk];
  const int mtile = blockIdx.x;
  if (mtile * 16 >= cnt) return;
  const bool fullTile = (mtile * 16 + 16) <= cnt;   // uniform: hot path has no masking

  const int   kTab[4]    = {1024, 256, 64, 16};
  const int   logKTab[4] = {10, 8, 6, 4};
  const int   lTab[4]    = {0, 20000, 40000, 200000};
  const float* embTab[4]  = {p.emb0, p.emb1, p.emb2, p.emb3};
  const float* projTab[4] = {p.proj0, p.proj1, p.proj2, p.proj3};

  const int Kd   = kTab[bk];
  const int logK = logKTab[bk];
  const int lo   = lTab[bk];
  gcfloat_p emb   = (gcfloat_p)embTab[bk];
  gcfloat_p proj  = (gcfloat_p)projTab[bk];
  gcint_p   inp   = (gcint_p)p.inp;
  gcint_p   lists = (gcint_p)p.lists;
  gfloat_p  out   = (gfloat_p)p.out;

  const int tid = threadIdx.x;

  // Stage the 16 token ids + gathered local row indices of this m-tile
  if (tid < 16) {
    int m = mtile * 16 + tid;
    int tok = (m < cnt) ? lists[bk * p.nTok + m] : -1;
    stok[tid] = tok;
    sloc[tid] = (tok >= 0) ? (inp[tok] - lo) : -1;
  }
  __syncthreads();

  // Gather A tile (16 x K) into LDS, coalesced along K; pad rows by 2 floats
  const int stride = Kd + 2;
  if (fullTile) {
    for (int i = tid; i < 16 * Kd; i += 256) {
      int r = i >> logK;
      int c = i & (Kd - 1);
      ldsA[r * stride + c] = emb[((size_t)sloc[r] << logK) + c];
    }
  } else {
    for (int i = tid; i < 16 * Kd; i += 256) {
      int r = i >> logK;
      int c = i & (Kd - 1);
      int li = sloc[r];
      ldsA[r * stride + c] = (li >= 0) ? emb[((size_t)li << logK) + c] : 0.0f;
    }
  }
  __syncthreads();

  const int lane = tid & 31;
  const int wave = tid >> 5;
  const int hi   = lane >> 4;      // which K-pair half (A/B frag layout)
  const int l16  = lane & 15;      // row (A) / col (B,C)

  const float* aRow = &ldsA[l16 * stride + 2 * hi];

  // Hoist per-row tokens / output row bases out of the column-tile loop
  int    tokr[8];
  size_t rowBase[8];
  #pragma unroll
  for (int r = 0; r < 8; ++r) {
    tokr[r]    = stok[r + 8 * hi];
    rowBase[r] = (size_t)(tokr[r] < 0 ? 0 : tokr[r]) * D_PROJ;
  }

  // Each wave owns 8 of the 64 16-col tiles; process them two at a time so each
  // A-fragment ds_load feeds 2 WMMAs and the loop has 2 independent accum chains.
  for (int j = 0; j < 4; ++j) {
    const int n0 = (wave + 16 * j) * 16 + l16;        // tile pair: w+16j, w+8+16j
    const int n1 = n0 + 8 * 16;
    gcfloat_p bPtr0 = proj + (size_t)(2 * hi) * D_PROJ + n0;  // rows k0, k0+1
    gcfloat_p bPtr1 = proj + (size_t)(2 * hi) * D_PROJ + n1;
    v8f c0 = {};
    v8f c1 = {};
    #pragma unroll 2
    for (int kk = 0; kk < Kd; kk += 4) {
      v2f a = *(const v2f*)(aRow + kk);               // A: 16 x K f32, 2 VGPRs/lane
      v2f b0, b1;
      b0.x = bPtr0[(size_t)kk * D_PROJ];              // B: K x N f32, 2 VGPRs/lane
      b0.y = bPtr0[(size_t)kk * D_PROJ + D_PROJ];
      b1.x = bPtr1[(size_t)kk * D_PROJ];
      b1.y = bPtr1[(size_t)kk * D_PROJ + D_PROJ];
      c0 = __builtin_amdgcn_wmma_f32_16x16x4_f32(
              false, a, false, b0, (short)0, c0, false, false);
      c1 = __builtin_amdgcn_wmma_f32_16x16x4_f32(
              false, a, false, b1, (short)0, c1, false, false);
    }
    // C layout: VGPR r -> row (r + 8*hi), col = lane&15 within the tile.
    // emb_scale = sqrt(1024) = 32; stream 128 MiB of output past L2 (NT stores).
    if (fullTile) {
      #pragma unroll
      for (int r = 0; r < 8; ++r) {
        __builtin_nontemporal_store(32.0f * c0[r], out + rowBase[r] + n0);
        __builtin_nontemporal_store(32.0f * c1[r], out + rowBase[r] + n1);
      }
    } else {
      #pragma unroll
      for (int r = 0; r < 8; ++r) {
        if (tokr[r] >= 0) {
          __builtin_nontemporal_store(32.0f * c0[r], out + rowBase[r] + n0);
          __builtin_nontemporal_store(32.0f * c1[r], out + rowBase[r] + n1);
        }
      }
    }
  }
}

extern "C" void kernel_launch(void* const* d_in, const int* in_sizes, int n_in,
                              void* d_out, int out_size, void* d_ws, size_t ws_size,
                              hipStream_t stream) {
  const int* inp = (const int*)d_in[0];
  const int nTok = in_sizes[0];

  int* counts = (int*)d_ws;
  int* lists  = counts + 8;   // keep lists 32B-aligned after 4 counters

  GemmParams p;
  p.emb0  = (const float*)d_in[1];
  p.emb1  = (const float*)d_in[2];
  p.emb2  = (const float*)d_in[3];
  p.emb3  = (const float*)d_in[4];
  p.proj0 = (const float*)d_in[5];
  p.proj1 = (const float*)d_in[6];
  p.proj2 = (const float*)d_in[7];
  p.proj3 = (const float*)d_in[8];
  p.inp    = inp;
  p.out    = (float*)d_out;
  p.counts = counts;
  p.lists  = lists;
  p.nTok   = nTok;

  init_counts_kernel<<<1, 32, 0, stream>>>(counts);
  classify_kernel<<<(nTok + 255) / 256, 256, 0, stream>>>(inp, nTok, counts, lists);

  const int mtiles = (nTok + 15) / 16;
  dim3 grid(mtiles, 4);
  const size_t shmem = (size_t)16 * (1024 + 2) * sizeof(float);  // worst case K=1024
  gemm_kernel<<<grid, 256, shmem, stream>>>(p);
}